// RelationalKENN_67594195305162
// MI455X (gfx1250) — compile-verified
//
#include <hip/hip_runtime.h>

#define N_UNARY   16
#define N_BINARY  4
#define N_NODES   100000
#define N_EDGES   1600000
#define N_CLAUSES 8

// ---------------------------------------------------------------------------
// Kernel A: unary knowledge enhancement.
//   u[n][:] = unary[n][:] + sum over clauses of w * sgn * softmax(sgn * x[idx])
//   U_IDX[i] = (i, i+1, i+2), U_SGN = (-1, +1, +1), i = 0..7
// Writes u to BOTH out_u (atomics from kernel B accumulate on top) and u_ws
// (clean copy that kernel B gathers from, so gathers never race the atomics).
// ---------------------------------------------------------------------------
__global__ __launch_bounds__(256)
void kenn_unary_kernel(const float* __restrict__ unary,
                       const float* __restrict__ ucw,
                       float* __restrict__ out_u,
                       float* __restrict__ u_ws)
{
    int n = blockIdx.x * blockDim.x + threadIdx.x;
    if (n >= N_NODES) return;

    const float4* src = reinterpret_cast<const float4*>(unary + (size_t)n * N_UNARY);
    float4 r0 = src[0], r1 = src[1], r2 = src[2], r3 = src[3];
    float x[16] = { r0.x, r0.y, r0.z, r0.w, r1.x, r1.y, r1.z, r1.w,
                    r2.x, r2.y, r2.z, r2.w, r3.x, r3.y, r3.z, r3.w };
    float d[16];
#pragma unroll
    for (int j = 0; j < 16; ++j) d[j] = 0.0f;

#pragma unroll
    for (int i = 0; i < N_CLAUSES; ++i) {
        float w  = ucw[i];
        float s0 = -x[i];          // sign -1 on literal 0
        float s1 =  x[i + 1];
        float s2 =  x[i + 2];
        float m  = fmaxf(s0, fmaxf(s1, s2));
        float e0 = __expf(s0 - m);
        float e1 = __expf(s1 - m);
        float e2 = __expf(s2 - m);
        float inv = __builtin_amdgcn_rcpf(e0 + e1 + e2);
        d[i]     -= w * e0 * inv;  // scatter back with sign -1
        d[i + 1] += w * e1 * inv;
        d[i + 2] += w * e2 * inv;
    }

    float u[16];
#pragma unroll
    for (int j = 0; j < 16; ++j) u[j] = x[j] + d[j];

    float4 o0 = make_float4(u[0],  u[1],  u[2],  u[3]);
    float4 o1 = make_float4(u[4],  u[5],  u[6],  u[7]);
    float4 o2 = make_float4(u[8],  u[9],  u[10], u[11]);
    float4 o3 = make_float4(u[12], u[13], u[14], u[15]);

    float4* dst0 = reinterpret_cast<float4*>(out_u + (size_t)n * N_UNARY);
    dst0[0] = o0; dst0[1] = o1; dst0[2] = o2; dst0[3] = o3;
    float4* dst1 = reinterpret_cast<float4*>(u_ws + (size_t)n * N_UNARY);
    dst1[0] = o0; dst1[1] = o1; dst1[2] = o2; dst1[3] = o3;
}

// ---------------------------------------------------------------------------
// Kernel B: binary knowledge enhancement over edges.
//   B_IDX[i] = (i, 32 + i%4, 16 + i), B_SGN = (-1, +1, +1)
//   joined col i      -> u[index1] col i   (0..7)
//   joined col 16+i   -> u[index2] col i   (0..7)
//   joined col 32+c   -> binary col c      (0..3)
// The two random 32B row gathers per edge go through the CDNA5 async path:
// global_load_async_to_lds_b128 (per-lane global addr -> per-lane LDS slot),
// overlapped with the coalesced binary load, then s_wait_asynccnt 0.
// ---------------------------------------------------------------------------
__global__ __launch_bounds__(256)
void kenn_edge_kernel(const float* __restrict__ binary,
                      const float* __restrict__ bcw,
                      const int* __restrict__ index1,
                      const int* __restrict__ index2,
                      const float* __restrict__ u_ws,
                      float* __restrict__ out_u,
                      float* __restrict__ out_b)
{
    __shared__ alignas(16) float shA[256 * 8];   // u[index1] cols 0..7 per lane
    __shared__ alignas(16) float shB[256 * 8];   // u[index2] cols 0..7 per lane

    int tid = threadIdx.x;
    long e = (long)blockIdx.x * blockDim.x + tid;
    if (e >= N_EDGES) return;

    int i1 = index1[e];
    int i2 = index2[e];

    const float* pA = u_ws + (size_t)i1 * N_UNARY;   // 32 bytes: cols 0..7
    const float* pB = u_ws + (size_t)i2 * N_UNARY;

    // Low 32 bits of the generic pointer to LDS == LDS byte offset (ISA 10.2).
    unsigned ldsA = (unsigned)(size_t)(void*)(&shA[tid * 8]);
    unsigned ldsB = (unsigned)(size_t)(void*)(&shB[tid * 8]);

    // 4 x 16B async gathers; INST_OFFSET applies to both LDS and global addr.
    asm volatile(
        "global_load_async_to_lds_b128 %0, %2, off\n\t"
        "global_load_async_to_lds_b128 %0, %2, off offset:16\n\t"
        "global_load_async_to_lds_b128 %1, %3, off\n\t"
        "global_load_async_to_lds_b128 %1, %3, off offset:16"
        :
        : "v"(ldsA), "v"(ldsB), "v"(pA), "v"(pB)
        : "memory");

    // Coalesced streaming load of this edge's binary row while gathers fly.
    float4 bin = *reinterpret_cast<const float4*>(binary + (size_t)e * N_BINARY);
    float bv[4] = { bin.x, bin.y, bin.z, bin.w };

    asm volatile("s_wait_asynccnt 0x0" ::: "memory");
    // Each lane only reads the LDS slots it targeted itself -> no barrier.

    float4 a0 = *reinterpret_cast<const float4*>(&shA[tid * 8]);
    float4 a1 = *reinterpret_cast<const float4*>(&shA[tid * 8 + 4]);
    float4 b0 = *reinterpret_cast<const float4*>(&shB[tid * 8]);
    float4 b1 = *reinterpret_cast<const float4*>(&shB[tid * 8 + 4]);
    float a[8] = { a0.x, a0.y, a0.z, a0.w, a1.x, a1.y, a1.z, a1.w };
    float b[8] = { b0.x, b0.y, b0.z, b0.w, b1.x, b1.y, b1.z, b1.w };

    float bd[4] = { 0.0f, 0.0f, 0.0f, 0.0f };
    float* du1 = out_u + (size_t)i1 * N_UNARY;
    float* du2 = out_u + (size_t)i2 * N_UNARY;

#pragma unroll
    for (int i = 0; i < N_CLAUSES; ++i) {
        float w  = bcw[i];
        float s0 = -a[i];          // joined col i,      sign -1
        float s1 =  bv[i & 3];     // joined col 32+i%4, sign +1
        float s2 =  b[i];          // joined col 16+i,   sign +1
        float m  = fmaxf(s0, fmaxf(s1, s2));
        float e0 = __expf(s0 - m);
        float e1 = __expf(s1 - m);
        float e2 = __expf(s2 - m);
        float inv = __builtin_amdgcn_rcpf(e0 + e1 + e2);
        atomicAdd(du1 + i, -w * e0 * inv);   // delta_up via index1
        bd[i & 3] += w * e1 * inv;           // binary delta (private)
        atomicAdd(du2 + i,  w * e2 * inv);   // delta_up via index2
    }

    float4 ob = make_float4(bv[0] + bd[0], bv[1] + bd[1],
                            bv[2] + bd[2], bv[3] + bd[3]);
    *reinterpret_cast<float4*>(out_b + (size_t)e * N_BINARY) = ob;
}

// ---------------------------------------------------------------------------
// kernel_launch
// ---------------------------------------------------------------------------
extern "C" void kernel_launch(void* const* d_in, const int* in_sizes, int n_in,
                              void* d_out, int out_size, void* d_ws, size_t ws_size,
                              hipStream_t stream)
{
    const float* unary  = (const float*)d_in[0];   // [N_NODES, 16]
    const float* binary = (const float*)d_in[1];   // [N_EDGES, 4]
    const float* ucw    = (const float*)d_in[2];   // [8]
    const float* bcw    = (const float*)d_in[3];   // [8]
    const int*   index1 = (const int*)d_in[4];     // [N_EDGES]
    const int*   index2 = (const int*)d_in[5];     // [N_EDGES]

    float* out_u = (float*)d_out;                              // [N_NODES, 16]
    float* out_b = (float*)d_out + (size_t)N_NODES * N_UNARY;  // [N_EDGES, 4]
    float* u_ws  = (float*)d_ws;                               // clean u, 6.4 MB

    kenn_unary_kernel<<<(N_NODES + 255) / 256, 256, 0, stream>>>(
        unary, ucw, out_u, u_ws);

    kenn_edge_kernel<<<(N_EDGES + 255) / 256, 256, 0, stream>>>(
        binary, bcw, index1, index2, u_ws, out_u, out_b);
}